// RandomMicEQ_146028888563
// MI455X (gfx1250) — compile-verified
//
#include <hip/hip_runtime.h>
#include <cmath>

#ifndef M_PI
#define M_PI 3.14159265358979323846
#endif

typedef float v2f __attribute__((ext_vector_type(2)));
typedef float v8f __attribute__((ext_vector_type(8)));

#define SEG   512                 // output samples per segment
#define WARM  496                 // overlap-save warmup (worst transient ~3e-7)
#define LEN   (SEG + WARM)        // 1008 samples processed per segment
#define LENP  1012                // LDS column stride; 1012 % 64 == 52 -> conflict-free column gathers
#define NCOL  16                  // WMMA N dimension = batch columns per wave
#define CHUNKS (LEN / 16)         // 63 sequential 16-sample blocks per segment
#define CSTOFF (NCOL * LENP)      // offset of the 80-float constant block in smem

struct FiltConsts {
    float b[3][3];   // normalized b0,b1,b2 per stage
    float a[3][2];   // normalized a1,a2  per stage
};

__global__ __launch_bounds__(32)
void biquad3_wmma(const float* __restrict__ x, float* __restrict__ y,
                  int T, int nseg, FiltConsts fc) {
    // 16 columns x (LEN+2) in-place working samples + per-stage constant tables
    __shared__ float smem[NCOL * LENP + 80];

    const int lane = threadIdx.x;        // 0..31, wave always fully active
    const int col  = lane & 15;
    const int hi   = lane >> 4;

    const int tile = blockIdx.x;
    const int rg   = tile / nseg;        // batch row group (0: rows 0-15, 1: rows 16-31)
    const int seg  = tile % nseg;
    const long row0   = (long)rg * 16;
    const int  tstart = seg * SEG - WARM - 2;

    // ---- stage-0 input: coalesced global -> LDS (zero-pad out of range) ----
    for (int c = 0; c < NCOL; ++c) {
        const float* src = x + (row0 + c) * (long)T;
        const int base = c * LENP;
        for (int i = lane; i < LEN + 2; i += 32) {
            const int t = tstart + i;
            smem[base + i] = (t >= 0 && t < T) ? src[t] : 0.0f;
        }
    }
    __syncthreads();

    for (int s = 0; s < 3; ++s) {
        // lane 0: build the block-recurrence constant tables in double.
        //   g[n]   = b0 h[n] + b1 h[n-1] + b2 h[n-2]   (full biquad impulse response)
        //   gu1[n] = b1 h[n] + b2 h[n-1]               (injection of u[-1])
        //   gu2[n] = b2 h[n]                           (injection of u[-2])
        //   hy1/hy2: homogeneous responses             (injection of y[-1], y[-2])
        if (lane == 0) {
            const double b0 = fc.b[s][0], b1 = fc.b[s][1], b2 = fc.b[s][2];
            const double a1 = fc.a[s][0], a2 = fc.a[s][1];
            double hp1 = 0.0, hp2 = 0.0;          // h[n-1], h[n-2]
            for (int n = 0; n < 16; ++n) {
                const double hn = (n == 0 ? 1.0 : 0.0) - a1 * hp1 - a2 * hp2;
                smem[CSTOFF +      n] = (float)(b0 * hn + b1 * hp1 + b2 * hp2);
                smem[CSTOFF + 16 + n] = (float)(b1 * hn + b2 * hp1);
                smem[CSTOFF + 32 + n] = (float)(b2 * hn);
                hp2 = hp1; hp1 = hn;
            }
            double p1 = 1.0, p2 = 0.0;
            for (int n = 0; n < 16; ++n) {
                const double hv = -a1 * p1 - a2 * p2;
                smem[CSTOFF + 48 + n] = (float)hv; p2 = p1; p1 = hv;
            }
            p1 = 0.0; p2 = 1.0;
            for (int n = 0; n < 16; ++n) {
                const double hv = -a1 * p1 - a2 * p2;
                smem[CSTOFF + 64 + n] = (float)hv; p2 = p1; p1 = hv;
            }
        }
        __syncthreads();

        // A fragments: G[m][k] = g[m-k] (m>=k); ISA 16x4 f32 A layout:
        // lane (m=col, hi): a.x = A[m][4kk+2hi], a.y = A[m][4kk+1+2hi]
        v2f afrag[4];
        #pragma unroll
        for (int kk = 0; kk < 4; ++kk) {
            const int kx = 4 * kk + 2 * hi;
            afrag[kk].x = (col >= kx)     ? smem[CSTOFF + col - kx]     : 0.0f;
            afrag[kk].y = (col >= kx + 1) ? smem[CSTOFF + col - kx - 1] : 0.0f;
        }
        // D-vgpr r holds row m = r + 8*hi
        float gu1r[8], gu2r[8], hy1r[8], hy2r[8];
        #pragma unroll
        for (int r = 0; r < 8; ++r) {
            gu1r[r] = smem[CSTOFF + 16 + 8 * hi + r];
            gu2r[r] = smem[CSTOFF + 32 + 8 * hi + r];
            hy1r[r] = smem[CSTOFF + 48 + 8 * hi + r];
            hy2r[r] = smem[CSTOFF + 64 + 8 * hi + r];
        }

        const int cb = col * LENP;
        float y_m1 = 0.0f, y_m2 = 0.0f;                    // unclamped IIR state
        float u_m1 = smem[cb + 1], u_m2 = smem[cb + 0];    // input history u[-1], u[-2]

        for (int ch = 0; ch < CHUNKS; ++ch) {
            const int tb = cb + 2 + ch * 16;

            // B fragments: raw input samples, uniform ds_load_b64 gathers
            v2f bfr[4];
            #pragma unroll
            for (int kk = 0; kk < 4; ++kk) {
                const int kx = 4 * kk + 2 * hi;
                bfr[kk].x = smem[tb + kx];
                bfr[kk].y = smem[tb + kx + 1];
            }

            // two independent accumulation chains over K
            const v8f z = {0.f,0.f,0.f,0.f,0.f,0.f,0.f,0.f};
            v8f d0 = __builtin_amdgcn_wmma_f32_16x16x4_f32(
                         false, afrag[0], false, bfr[0], (short)0, z, false, false);
            v8f d1 = __builtin_amdgcn_wmma_f32_16x16x4_f32(
                         false, afrag[1], false, bfr[1], (short)0, z, false, false);
            d0 = __builtin_amdgcn_wmma_f32_16x16x4_f32(
                         false, afrag[2], false, bfr[2], (short)0, d0, false, false);
            d1 = __builtin_amdgcn_wmma_f32_16x16x4_f32(
                         false, afrag[3], false, bfr[3], (short)0, d1, false, false);

            // merge chains + inject boundary terms
            v8f d;
            #pragma unroll
            for (int r = 0; r < 8; ++r)
                d[r] = d0[r] + d1[r]
                     + gu1r[r] * u_m1 + gu2r[r] * u_m2
                     + hy1r[r] * y_m1 + hy2r[r] * y_m2;

            // next-chunk state: rows 15,14 (unclamped), broadcast per column
            const float nm1 = __shfl(d[7], 16 + col, 32);
            const float nm2 = __shfl(d[6], 16 + col, 32);
            y_m1 = nm1; y_m2 = nm2;

            // refresh input history (uniform loads) before the in-place overwrite
            u_m2 = smem[tb + 14];
            u_m1 = smem[tb + 15];

            // clamp and store stage output in place (next stage's input)
            #pragma unroll
            for (int r = 0; r < 8; ++r) {
                float v = d[r];
                v = fminf(fmaxf(v, -1.0f), 1.0f);
                smem[tb + r + 8 * hi] = v;
            }
        }
        __syncthreads();
    }

    // ---- write back only the non-warmup region, coalesced ----
    const int t0 = seg * SEG;
    for (int c = 0; c < NCOL; ++c) {
        float* dst = y + (row0 + c) * (long)T;
        const int base = c * LENP + 2 + WARM;
        for (int i = lane; i < SEG; i += 32) {
            const int t = t0 + i;
            if (t < T) dst[t] = smem[base + i];
        }
    }
}

// ---------------- host side ----------------
static void norm3(const double* b, const double* a, float* bo, float* ao) {
    bo[0] = (float)(b[0] / a[0]); bo[1] = (float)(b[1] / a[0]); bo[2] = (float)(b[2] / a[0]);
    ao[0] = (float)(a[1] / a[0]); ao[1] = (float)(a[2] / a[0]);
}

extern "C" void kernel_launch(void* const* d_in, const int* in_sizes, int n_in,
                              void* d_out, int out_size, void* d_ws, size_t ws_size,
                              hipStream_t stream) {
    (void)n_in; (void)d_ws; (void)ws_size;
    const double SR = 44100.0;

    FiltConsts fc;
    {   // equalizer_biquad(1000 Hz, +6 dB, Q=1)
        const double w0 = 2.0 * M_PI * 1000.0 / SR;
        const double alpha = sin(w0) / (2.0 * 1.0);
        const double A = pow(10.0, 6.0 / 40.0);
        const double b[3] = {1.0 + alpha * A, -2.0 * cos(w0), 1.0 - alpha * A};
        const double a[3] = {1.0 + alpha / A, -2.0 * cos(w0), 1.0 - alpha / A};
        norm3(b, a, fc.b[0], fc.a[0]);
    }
    {   // lowpass_biquad(8000 Hz, Q=1/sqrt(2))
        const double Q = 0.7071067811865476;
        const double w0 = 2.0 * M_PI * 8000.0 / SR;
        const double alpha = sin(w0) / (2.0 * Q);
        const double c = cos(w0);
        const double b[3] = {(1.0 - c) / 2.0, 1.0 - c, (1.0 - c) / 2.0};
        const double a[3] = {1.0 + alpha, -2.0 * c, 1.0 - alpha};
        norm3(b, a, fc.b[1], fc.a[1]);
    }
    {   // highpass_biquad(300 Hz, Q=1/sqrt(2))
        const double Q = 0.7071067811865476;
        const double w0 = 2.0 * M_PI * 300.0 / SR;
        const double alpha = sin(w0) / (2.0 * Q);
        const double c = cos(w0);
        const double b[3] = {(1.0 + c) / 2.0, -(1.0 + c), (1.0 + c) / 2.0};
        const double a[3] = {1.0 + alpha, -2.0 * c, 1.0 - alpha};
        norm3(b, a, fc.b[2], fc.a[2]);
    }

    const float* x = (const float*)d_in[0];
    float* out = (float*)d_out;
    const int B = 32;
    const int T = (in_sizes && in_sizes[0] > 0) ? in_sizes[0] / B : out_size / B;
    const int nseg = (T + SEG - 1) / SEG;
    const int blocks = 2 * nseg;   // 2 row-groups of 16 batch rows each

    biquad3_wmma<<<blocks, 32, 0, stream>>>(x, out, T, nseg, fc);
}